// OnlyMarkovConvolution_27410481283764
// MI455X (gfx1250) — compile-verified
//
#include <hip/hip_runtime.h>
#include <stdint.h>

// --- problem constants (match reference) ---
#define NCOLS       2048
#define ROWS_PER_WG 8
#define THREADS     256
#define NWAVES      (THREADS / 32)
#define MASK_FILL   (-1.0e8f)

typedef __attribute__((ext_vector_type(4))) unsigned int v4u;
typedef __attribute__((ext_vector_type(8))) int          v8i;
typedef __attribute__((ext_vector_type(4))) int          v4i;

// Issue a TDM DMA of one 2048-float row from global memory into LDS.
// D# group0: count=1 | lds_addr | global_addr[56:0] | type=2 (bits 127:126).
// D# group1: data_size=4B (code 2), tensor_dim0=2048, tensor_dim1=1,
//            tile 2048x1, tensor_dim0_stride=2048. Groups 2/3 (+extra) = 0.
__device__ __forceinline__ void tdm_load_row(const float* gsrc, unsigned lds_off) {
  uint64_t ga = (uint64_t)(uintptr_t)gsrc;
  v4u g0;
  g0[0] = 1u;                                               // count=1, user mode
  g0[1] = lds_off;                                          // LDS byte address
  g0[2] = (unsigned)ga;                                     // global_addr[31:0]
  g0[3] = (unsigned)((ga >> 32) & 0x1FFFFFFull) | (2u << 30); // addr[56:32] | type=2
  v8i g1;
  g1[0] = 0x00020000;              // workgroup_mask=0, data_size=2 (4 bytes)
  g1[1] = (int)(NCOLS << 16);      // tensor_dim0[15:0] in [31:16]
  g1[2] = (int)(1u << 16);         // tensor_dim0[31:16]=0, tensor_dim1 lo = 1
  g1[3] = (int)(NCOLS << 16);      // tensor_dim1 hi = 0, tile_dim0 = 2048
  g1[4] = 1;                       // tile_dim1 = 1, tile_dim2 = 0
  g1[5] = NCOLS;                   // tensor_dim0_stride lo32 (in elements)
  g1[6] = 0;                       // stride0 hi, stride1 lo
  g1[7] = 0;
  v4i z4 = {0, 0, 0, 0};
  v8i z8 = {0, 0, 0, 0, 0, 0, 0, 0};
  // 6-arg toolchain variant: (g0, g1, g2, g3, g4, cpol)
  __builtin_amdgcn_tensor_load_to_lds(g0, g1, z4, z4, z8, /*cpol=*/0);
}

__global__ void __launch_bounds__(THREADS)
markov_logsoftmax_kernel(const float* __restrict__ markov,
                         const int*   __restrict__ mask,
                         const float* __restrict__ conv_w,
                         const float* __restrict__ conv_b,
                         float*       __restrict__ out) {
  __shared__ float buf[2][NCOLS];     // ping-pong row buffers (TDM targets)
  __shared__ float red_m[NWAVES];
  __shared__ float red_s[NWAVES];

  const int tid  = threadIdx.x;
  const int wave = tid >> 5;
  const int lane = tid & 31;
  const int row0 = blockIdx.x * ROWS_PER_WG;
  const int c0   = tid * 4;           // first float4 chunk (cols [c0, c0+4))
  const int c1   = c0 + NCOLS / 2;    // second float4 chunk

  // Per-column affine params: fixed per thread across all rows -> hoist.
  const float4 w0 = *reinterpret_cast<const float4*>(conv_w + c0);
  const float4 w1 = *reinterpret_cast<const float4*>(conv_w + c1);
  const float4 b0 = *reinterpret_cast<const float4*>(conv_b + c0);
  const float4 b1 = *reinterpret_cast<const float4*>(conv_b + c1);

  const unsigned lds0 = (unsigned)(uintptr_t)&buf[0][0];
  const unsigned lds1 = (unsigned)(uintptr_t)&buf[1][0];

  // Prime the pipeline: wave 0 DMAs row 0 into buf[0].
  if (wave == 0) {
    tdm_load_row(markov + (size_t)row0 * NCOLS, lds0);
  }

  for (int t = 0; t < ROWS_PER_WG; ++t) {
    const int row = row0 + t;

    if (wave == 0) {
      if (t + 1 < ROWS_PER_WG) {
        // Issue next row's DMA, then wait so the *previous* one (row t) is done.
        tdm_load_row(markov + (size_t)(row + 1) * NCOLS,
                     ((t + 1) & 1) ? lds1 : lds0);
        __builtin_amdgcn_s_wait_tensorcnt((short)1);
      } else {
        __builtin_amdgcn_s_wait_tensorcnt((short)0);
      }
    }
    __syncthreads();   // row t resident in LDS for all waves

    const float* rb = buf[t & 1];
    const float4 m0 = *reinterpret_cast<const float4*>(rb + c0);  // ds_load_b128
    const float4 m1 = *reinterpret_cast<const float4*>(rb + c1);

    // Prefetch next row's mask while we reduce this row.
    if (t + 1 < ROWS_PER_WG) {
      __builtin_prefetch(mask + (size_t)(row + 1) * NCOLS + c0, 0, 1);
    }

    float v[8];
    v[0] = fmaf(w0.x, m0.x, b0.x); v[1] = fmaf(w0.y, m0.y, b0.y);
    v[2] = fmaf(w0.z, m0.z, b0.z); v[3] = fmaf(w0.w, m0.w, b0.w);
    v[4] = fmaf(w1.x, m1.x, b1.x); v[5] = fmaf(w1.y, m1.y, b1.y);
    v[6] = fmaf(w1.z, m1.z, b1.z); v[7] = fmaf(w1.w, m1.w, b1.w);

    // Thread-local max + sum(exp).
    float m = v[0];
#pragma unroll
    for (int i = 1; i < 8; ++i) m = fmaxf(m, v[i]);
    float s = 0.0f;
#pragma unroll
    for (int i = 0; i < 8; ++i) s += __expf(v[i] - m);

    // Wave32 (m,s) merge via xor-shuffle butterfly.
#pragma unroll
    for (int off = 16; off > 0; off >>= 1) {
      const float om = __shfl_xor(m, off, 32);
      const float os = __shfl_xor(s, off, 32);
      const float nm = fmaxf(m, om);
      s = s * __expf(m - nm) + os * __expf(om - nm);
      m = nm;
    }
    if (lane == 0) { red_m[wave] = m; red_s[wave] = s; }
    __syncthreads();

    // Merge the 8 per-wave partials (cheap, every thread redundantly).
    float M = red_m[0], S = red_s[0];
#pragma unroll
    for (int i = 1; i < NWAVES; ++i) {
      const float om = red_m[i], os = red_s[i];
      const float nm = fmaxf(M, om);
      S = S * __expf(M - nm) + os * __expf(om - nm);
      M = nm;
    }
    const float offset = M + __logf(S);   // log_softmax shift

    // Masked write-out: b128 mask loads, b128 stores.
    const size_t base = (size_t)row * NCOLS;
    const v4i k0 = *reinterpret_cast<const v4i*>(mask + base + c0);
    const v4i k1 = *reinterpret_cast<const v4i*>(mask + base + c1);

    float4 o0, o1;
    o0.x = k0[0] ? MASK_FILL : v[0] - offset;
    o0.y = k0[1] ? MASK_FILL : v[1] - offset;
    o0.z = k0[2] ? MASK_FILL : v[2] - offset;
    o0.w = k0[3] ? MASK_FILL : v[3] - offset;
    o1.x = k1[0] ? MASK_FILL : v[4] - offset;
    o1.y = k1[1] ? MASK_FILL : v[5] - offset;
    o1.z = k1[2] ? MASK_FILL : v[6] - offset;
    o1.w = k1[3] ? MASK_FILL : v[7] - offset;
    *reinterpret_cast<float4*>(out + base + c0) = o0;
    *reinterpret_cast<float4*>(out + base + c1) = o1;

    __syncthreads();   // protect buf[t&1] and red_* before reuse
  }
}

extern "C" void kernel_launch(void* const* d_in, const int* in_sizes, int n_in,
                              void* d_out, int out_size, void* d_ws, size_t ws_size,
                              hipStream_t stream) {
  // setup_inputs order: x(0), x_dist(1), x_features(2), x_markov(3),
  //                     x_week(4), x_mask(5), conv_w(6), conv_b(7)
  const float* markov = (const float*)d_in[3];
  const int*   mask   = (const int*)  d_in[5];
  const float* conv_w = (const float*)d_in[6];
  const float* conv_b = (const float*)d_in[7];
  float*       out    = (float*)d_out;

  const int B = in_sizes[4];                 // x_week has B elements (16384)
  dim3 grid(B / ROWS_PER_WG), block(THREADS);
  hipLaunchKernelGGL(markov_logsoftmax_kernel, grid, block, 0, stream,
                     markov, mask, conv_w, conv_b, out);
  (void)n_in; (void)out_size; (void)d_ws; (void)ws_size;
}